// GraphSAGE_15556371546548
// MI455X (gfx1250) — compile-verified
//
#include <hip/hip_runtime.h>

typedef float v2f __attribute__((ext_vector_type(2)));
typedef float v8f __attribute__((ext_vector_type(8)));

#define DIN 128  // K dimension for both layers (d_in == d_hid == 128)

// ---------------------------------------------------------------------------
// Degree: one thread per edge, atomic f32 add into deg[dst].
// ---------------------------------------------------------------------------
__global__ __launch_bounds__(256) void degree_kernel(
    const int* __restrict__ ei,   // [2, E] row-major: ei[e]=dst, ei[E+e]=src
    float* __restrict__ deg,
    int nedges)
{
    int e = blockIdx.x * blockDim.x + threadIdx.x;
    if (e < nedges) atomicAdd(&deg[ei[e]], 1.0f);
}

// ---------------------------------------------------------------------------
// Scatter-add: 32 threads per edge, each thread moves a float4 chunk.
// Coalesced 512B gather of feat[src], atomic adds into agg[dst] row.
// ---------------------------------------------------------------------------
__global__ __launch_bounds__(256) void scatter_add_kernel(
    const float* __restrict__ feat,   // [N, 128]
    const int*   __restrict__ ei,     // [2, E]
    float*       __restrict__ agg,    // [N, 128] (pre-zeroed)
    int nedges)
{
    long long tid = (long long)blockIdx.x * blockDim.x + threadIdx.x;
    long long total = (long long)nedges * 32;   // 128 floats / 4 per thread
    if (tid >= total) return;
    int e     = (int)(tid >> 5);
    int chunk = (int)(tid & 31);
    int dst = ei[e];
    int src = ei[nedges + e];
    const float4 v = *(const float4*)(feat + (size_t)src * DIN + chunk * 4);
    float* p = agg + (size_t)dst * DIN + chunk * 4;
    atomicAdd(p + 0, v.x);
    atomicAdd(p + 1, v.y);
    atomicAdd(p + 2, v.z);
    atomicAdd(p + 3, v.w);
}

// ---------------------------------------------------------------------------
// Fused SAGE layer GEMM via V_WMMA_F32_16X16X4_F32:
//   out = act( (agg * 1/max(deg,1)) @ Wl + xin @ Wr + bias )
// One wave -> one 16x16 output tile; 8 waves/block cover 8 row-tiles.
// Fragment layout per ISA 7.12.2 (f32 16x16x4): A: lane=m+16*(k/2), vgpr=k%2;
// B: lane=n+16*(k/2), vgpr=k%2;  C/D: lane=n (lo16)|n (hi16 -> m+8), vgpr=m%8.
// ---------------------------------------------------------------------------
template <int DOUT, bool DO_SELU>
__global__ __launch_bounds__(256) void sage_gemm_wmma(
    const float* __restrict__ agg,   // [N, DIN]
    const float* __restrict__ xin,   // [N, DIN]
    const float* __restrict__ deg,   // [N]
    const float* __restrict__ Wl,    // [DIN, DOUT] row-major
    const float* __restrict__ Wr,    // [DIN, DOUT] row-major
    const float* __restrict__ bias,  // [DOUT]
    float*       __restrict__ out,   // [N, DOUT]
    int nrows)
{
    const int lane  = threadIdx.x & 31;
    const int wave  = threadIdx.x >> 5;
    const int mtile = blockIdx.x * 8 + wave;     // 16-row tile index
    const int ntile = blockIdx.y;                // 16-col tile index
    if (mtile * 16 >= nrows) return;             // uniform per wave: EXEC stays all-1s

    const int half = lane >> 4;                  // 0 or 1
    const int lr   = lane & 15;
    const int m    = mtile * 16 + lr;            // A row handled by this lane
    const int n    = ntile * 16 + lr;            // B/C column handled by this lane

    const float scale = 1.0f / fmaxf(deg[m], 1.0f);
    const float* aggRow = agg + (size_t)m * DIN;
    const float* xRow   = xin + (size_t)m * DIN;

    v8f c = {};
#pragma unroll 4
    for (int k0 = 0; k0 < DIN; k0 += 4) {
        const int ka = k0 + 2 * half;            // lane's k pair: ka, ka+1
        v2f aAgg, aX, bL, bR;
        aAgg.x = aggRow[ka]     * scale;
        aAgg.y = aggRow[ka + 1] * scale;
        aX.x   = xRow[ka];
        aX.y   = xRow[ka + 1];
        bL.x   = Wl[(size_t)ka * DOUT + n];
        bL.y   = Wl[(size_t)(ka + 1) * DOUT + n];
        bR.x   = Wr[(size_t)ka * DOUT + n];
        bR.y   = Wr[(size_t)(ka + 1) * DOUT + n];
        // D = A*B + C  (f32, exact accumulate)
        c = __builtin_amdgcn_wmma_f32_16x16x4_f32(
                false, aAgg, false, bL, (short)0, c, false, false);
        c = __builtin_amdgcn_wmma_f32_16x16x4_f32(
                false, aX,   false, bR, (short)0, c, false, false);
    }

    const float bn = bias[n];
    float* outBase = out + (size_t)(mtile * 16) * DOUT + (size_t)ntile * 16 + lr;
#pragma unroll
    for (int r = 0; r < 8; ++r) {
        const int mrow = r + 8 * half;           // row within the 16x16 tile
        float v = c[r] + bn;
        if (DO_SELU) {
            const float alpha = 1.6732632423543772f;
            const float sc    = 1.0507009873554805f;
            v = (v > 0.0f) ? sc * v : sc * alpha * (expf(v) - 1.0f);
        }
        outBase[(size_t)mrow * DOUT] = v;
    }
}

// ---------------------------------------------------------------------------
// Launch: deg -> scatter(x) -> gemm1(+selu) -> scatter(h) -> gemm2
// Inputs: 0:x 1:edge_index 2:W1_l 3:W1_r 4:b1 5:W2_l 6:W2_r 7:b2
// ---------------------------------------------------------------------------
extern "C" void kernel_launch(void* const* d_in, const int* in_sizes, int n_in,
                              void* d_out, int out_size, void* d_ws, size_t ws_size,
                              hipStream_t stream)
{
    const float* x   = (const float*)d_in[0];
    const int*   ei  = (const int*)d_in[1];
    const float* W1l = (const float*)d_in[2];
    const float* W1r = (const float*)d_in[3];
    const float* b1  = (const float*)d_in[4];
    const float* W2l = (const float*)d_in[5];
    const float* W2r = (const float*)d_in[6];
    const float* b2  = (const float*)d_in[7];
    float* out = (float*)d_out;

    const int N  = in_sizes[0] / DIN;        // 100000
    const int E  = in_sizes[1] / 2;          // 1600000
    const int DO = in_sizes[5] / DIN;        // 64

    // Workspace: deg [N] | agg [N,128] (reused both layers) | h [N,128]
    char* ws = (char*)d_ws;
    const size_t degBytes = ((size_t)N * sizeof(float) + 511) & ~(size_t)511;
    const size_t aggBytes = ((size_t)N * DIN * sizeof(float) + 511) & ~(size_t)511;
    float* deg = (float*)ws;
    float* agg = (float*)(ws + degBytes);
    float* h   = (float*)(ws + degBytes + aggBytes);

    const int mtiles   = (N + 15) / 16;      // N is a multiple of 16 here
    const int gemmGx   = (mtiles + 7) / 8;   // 8 waves (16-row tiles) per block
    const long long scatThreads = (long long)E * 32;
    const int scatBlocks = (int)((scatThreads + 255) / 256);

    // ---- Layer 1 ----
    hipMemsetAsync(deg, 0, (size_t)N * sizeof(float), stream);
    hipMemsetAsync(agg, 0, (size_t)N * DIN * sizeof(float), stream);
    degree_kernel<<<(E + 255) / 256, 256, 0, stream>>>(ei, deg, E);
    scatter_add_kernel<<<scatBlocks, 256, 0, stream>>>(x, ei, agg, E);
    sage_gemm_wmma<128, true><<<dim3(gemmGx, 128 / 16), 256, 0, stream>>>(
        agg, x, deg, W1l, W1r, b1, h, N);

    // ---- Layer 2 ----
    hipMemsetAsync(agg, 0, (size_t)N * DIN * sizeof(float), stream);
    scatter_add_kernel<<<scatBlocks, 256, 0, stream>>>(h, ei, agg, E);
    sage_gemm_wmma<64, false><<<dim3(gemmGx, 64 / 16), 256, 0, stream>>>(
        agg, h, deg, W2l, W2r, b2, out, N);
}